// Encoder_77146202571148
// MI455X (gfx1250) — compile-verified
//
#include <hip/hip_runtime.h>
#include <hip/hip_bf16.h>
#include <cfloat>

typedef __attribute__((ext_vector_type(2))) float v2f;
typedef __attribute__((ext_vector_type(8))) float v8f;

#define HN 64
#define HE 16
#define LROWS 144          // 2*HN + HE rows in attn_w
#define NEG_SLOPE 0.2f
#define BN_EPS 1e-5f
#define BN_ROWS 400        // rows per BN partial block

__device__ __forceinline__ void atomicMaxF(float* addr, float val) {
    int* ai = (int*)addr;
    int old = *ai;
    while (__int_as_float(old) < val) {
        int assumed = old;
        old = atomicCAS(ai, assumed, __float_as_int(val));
        if (old == assumed) break;
    }
}

// h0 = x @ fc_node_w + fc_node_b   ([N,3]@[3,64])
__global__ void node_enc(const float* __restrict__ x, const float* __restrict__ W,
                         const float* __restrict__ b, float* __restrict__ h, int N) {
    int idx = blockIdx.x * 256 + threadIdx.x;
    if (idx >= N * HN) return;
    int n = idx >> 6, c = idx & 63;
    h[idx] = b[c] + x[n * 3 + 0] * W[c] + x[n * 3 + 1] * W[HN + c] + x[n * 3 + 2] * W[2 * HN + c];
}

// per-block partial (sum, sumsq) per channel
__global__ void bn_partial(const float* __restrict__ h, float* __restrict__ part, int N) {
    __shared__ float ssum[256], ssq[256];
    int c = threadIdx.x & 63, r4 = threadIdx.x >> 6;
    int rbeg = blockIdx.x * BN_ROWS;
    int rend = min(N, rbeg + BN_ROWS);
    float s = 0.f, s2 = 0.f;
    for (int r = rbeg + r4; r < rend; r += 4) {
        float v = h[r * HN + c];
        s += v; s2 += v * v;
    }
    ssum[threadIdx.x] = s; ssq[threadIdx.x] = s2;
    __syncthreads();
    if (r4 == 0) {
        for (int i = 1; i < 4; ++i) { s += ssum[i * 64 + c]; s2 += ssq[i * 64 + c]; }
        part[blockIdx.x * 128 + c] = s;
        part[blockIdx.x * 128 + 64 + c] = s2;
    }
}

// deterministic fixed-order combine -> scale/shift  (64 threads)
__global__ void bn_finalize(const float* __restrict__ part, int nblk,
                            const float* __restrict__ gamma, const float* __restrict__ beta,
                            float* __restrict__ scale, float* __restrict__ shift, int N) {
    int c = threadIdx.x;
    float s = 0.f, s2 = 0.f;
    for (int b = 0; b < nblk; ++b) { s += part[b * 128 + c]; s2 += part[b * 128 + 64 + c]; }
    float mean = s / (float)N;
    float var  = s2 / (float)N - mean * mean;
    float sc = rsqrtf(var + BN_EPS) * gamma[c];
    scale[c] = sc;
    shift[c] = beta[c] - mean * sc;
}

__global__ void bn_apply(float* __restrict__ h, const float* __restrict__ scale,
                         const float* __restrict__ shift, int total) {
    int idx = blockIdx.x * 256 + threadIdx.x;
    if (idx >= total) return;
    int c = idx & 63;
    h[idx] = h[idx] * scale[c] + shift[c];
}

// rank-1 edge-feature fold: u = fc_edge_w @ W_e,  v = fc_edge_b @ W_e + attn_b  (64 threads)
__global__ void uv_kernel(const float* __restrict__ attn_w, const float* __restrict__ attn_b,
                          const float* __restrict__ few, const float* __restrict__ feb,
                          int l, float* __restrict__ u, float* __restrict__ v) {
    int c = threadIdx.x;
    const float* We = attn_w + (size_t)l * LROWS * HN + 2 * HN * HN;  // rows 128..143
    float uu = 0.f, vv = 0.f;
    for (int j = 0; j < HE; ++j) {
        float w = We[j * HN + c];
        uu += few[j] * w;
        vv += feb[j] * w;
    }
    u[c] = uu;
    v[c] = vv + attn_b[l * HN + c];
}

// D[nrows,64] = A[nrows,64] @ B[64x64 view, row stride HN] (+bias), stored at ldc
// One wave per 16x16 tile via V_WMMA_F32_16X16X4_F32 (exact fp32).
__global__ void gemm64_wmma(const float* __restrict__ A, const float* __restrict__ B,
                            const float* __restrict__ bias, float* __restrict__ C,
                            int ldc, int nrows) {
    int wave = threadIdx.x >> 5;
    int lane = threadIdx.x & 31;
    int row0 = (blockIdx.x * 2 + (wave >> 2)) * 16;
    int col0 = (wave & 3) * 16;
    if (row0 >= nrows) return;                 // wave-uniform; EXEC stays full
    int m = lane & 15;                          // M for A-frag, N for B-frag/store
    int g = lane >> 4;                          // lane group
    v8f acc = {0.f, 0.f, 0.f, 0.f, 0.f, 0.f, 0.f, 0.f};
    const float* arow = A + (row0 + m) * HN + 2 * g;
    const float* bp   = B + 2 * g * HN + col0 + m;
#pragma unroll
    for (int k = 0; k < HN; k += 4) {
        v2f af = {arow[k], arow[k + 1]};                 // A[m][k+2g], A[m][k+2g+1]
        v2f bf = {bp[k * HN], bp[k * HN + HN]};          // B[k+2g][n], B[k+2g+1][n]
        acc = __builtin_amdgcn_wmma_f32_16x16x4_f32(false, af, false, bf,
                                                    (short)0, acc, false, false);
    }
    float bv = bias ? bias[col0 + m] : 0.f;
    int rbase = row0 + 8 * g;
#pragma unroll
    for (int i = 0; i < 8; ++i)
        C[(rbase + i) * ldc + col0 + m] = acc[i] + bv;
}

__global__ void init_layer(float* __restrict__ amax, float* __restrict__ denom,
                           float* __restrict__ hout, int total) {
    int idx = blockIdx.x * 256 + threadIdx.x;
    if (idx >= total) return;
    amax[idx] = -FLT_MAX;
    denom[idx] = 0.f;
    hout[idx] = 0.f;
}

__device__ __forceinline__ float edge_alpha(const float* __restrict__ a, int s, int d, int c,
                                            float eaw, const float* __restrict__ u,
                                            const float* __restrict__ v) {
    float al = a[d * 128 + c] + a[s * 128 + 64 + c] + eaw * u[c] + v[c];
    return al > 0.f ? al : NEG_SLOPE * al;     // leaky_relu
}

__global__ void edge_pass_max(const int* __restrict__ ei, const float* __restrict__ ea,
                              const float* __restrict__ a, const float* __restrict__ u,
                              const float* __restrict__ v, float* __restrict__ amax, int E) {
    int idx = blockIdx.x * 256 + threadIdx.x;
    int k = idx >> 6, c = idx & 63;
    if (k >= E) return;
    int s = ei[k], d = ei[E + k];
    float al = edge_alpha(a, s, d, c, ea[k], u, v);
    atomicMaxF(&amax[d * HN + c], al);
}

__global__ void edge_pass_sum(const int* __restrict__ ei, const float* __restrict__ ea,
                              const float* __restrict__ a, const float* __restrict__ u,
                              const float* __restrict__ v, const float* __restrict__ amax,
                              float* __restrict__ denom, int E) {
    int idx = blockIdx.x * 256 + threadIdx.x;
    int k = idx >> 6, c = idx & 63;
    if (k >= E) return;
    int s = ei[k], d = ei[E + k];
    float al = edge_alpha(a, s, d, c, ea[k], u, v);
    float ex = __expf(al - amax[d * HN + c]);
    atomicAdd(&denom[d * HN + c], ex);
}

__global__ void edge_pass_agg(const int* __restrict__ ei, const float* __restrict__ ea,
                              const float* __restrict__ a, const float* __restrict__ u,
                              const float* __restrict__ v, const float* __restrict__ amax,
                              const float* __restrict__ denom, const float* __restrict__ hl,
                              float* __restrict__ hout, int E) {
    int idx = blockIdx.x * 256 + threadIdx.x;
    int k = idx >> 6, c = idx & 63;
    if (k >= E) return;
    int s = ei[k], d = ei[E + k];
    float al = edge_alpha(a, s, d, c, ea[k], u, v);
    float ex = __expf(al - amax[d * HN + c]);
    float w = ex / (denom[d * HN + c] + 1e-16f);
    atomicAdd(&hout[d * HN + c], w * hl[s * HN + c]);
}

extern "C" void kernel_launch(void* const* d_in, const int* in_sizes, int n_in,
                              void* d_out, int out_size, void* d_ws, size_t ws_size,
                              hipStream_t stream) {
    const float* x    = (const float*)d_in[0];
    const float* ea   = (const float*)d_in[1];
    const int*   ei   = (const int*)d_in[2];
    const float* fnw  = (const float*)d_in[3];
    const float* fnb  = (const float*)d_in[4];
    const float* few  = (const float*)d_in[5];
    const float* feb  = (const float*)d_in[6];
    const float* gamma = (const float*)d_in[7];
    const float* beta  = (const float*)d_in[8];
    const float* fcw  = (const float*)d_in[9];
    const float* fcb  = (const float*)d_in[10];
    const float* aw   = (const float*)d_in[11];
    const float* ab   = (const float*)d_in[12];

    const int N = in_sizes[0] / 3;
    const int E = in_sizes[1];
    const int N64 = N * HN;

    float* ws    = (float*)d_ws;
    float* bufA  = ws;               // [N,64] h ping
    float* hl    = bufA + N64;       // [N,64]
    float* amax  = hl + N64;         // [N,64]
    float* denom = amax + N64;       // [N,64]
    float* amat  = denom + N64;      // [N,128] = [a_i | a_j]
    float* uvec  = amat + (size_t)N * 128;
    float* vvec  = uvec + 64;
    float* scale = vvec + 64;
    float* shift = scale + 64;
    float* part  = shift + 64;       // BN partials

    float* hout_final = (float*)d_out;

    const int elemGrid = (N64 + 255) / 256;
    const int edgeGrid = ((E * 64) + 255) / 256;
    const int gemmGrid = (N + 31) / 32;
    const int bnBlocks = (N + BN_ROWS - 1) / BN_ROWS;

    // --- encoder + batchnorm ---
    node_enc<<<elemGrid, 256, 0, stream>>>(x, fnw, fnb, bufA, N);
    bn_partial<<<bnBlocks, 256, 0, stream>>>(bufA, part, N);
    bn_finalize<<<1, 64, 0, stream>>>(part, bnBlocks, gamma, beta, scale, shift, N);
    bn_apply<<<elemGrid, 256, 0, stream>>>(bufA, scale, shift, N64);

    // --- 3 conv layers, ping-pong h between bufA and d_out ---
    float* h_in = bufA;
    for (int l = 0; l < 3; ++l) {
        float* h_out = (l & 1) ? bufA : hout_final;   // l=0 -> d_out, l=1 -> bufA, l=2 -> d_out
        const float* Wl  = fcw + (size_t)l * HN * HN;
        const float* bl  = fcb + (size_t)l * HN;
        const float* AWl = aw + (size_t)l * LROWS * HN;

        // hl = h_in @ fc_w[l] + fc_b[l]
        gemm64_wmma<<<gemmGrid, 256, 0, stream>>>(h_in, Wl, bl, hl, HN, N);
        // a_i = hl @ W_i  (rows 0..63)   -> amat[:,0:64]
        gemm64_wmma<<<gemmGrid, 256, 0, stream>>>(hl, AWl, nullptr, amat, 128, N);
        // a_j = hl @ W_j  (rows 64..127) -> amat[:,64:128]
        gemm64_wmma<<<gemmGrid, 256, 0, stream>>>(hl, AWl + HN * HN, nullptr, amat + HN, 128, N);

        uv_kernel<<<1, 64, 0, stream>>>(aw, ab, few, feb, l, uvec, vvec);
        init_layer<<<elemGrid, 256, 0, stream>>>(amax, denom, h_out, N64);

        edge_pass_max<<<edgeGrid, 256, 0, stream>>>(ei, ea, amat, uvec, vvec, amax, E);
        edge_pass_sum<<<edgeGrid, 256, 0, stream>>>(ei, ea, amat, uvec, vvec, amax, denom, E);
        edge_pass_agg<<<edgeGrid, 256, 0, stream>>>(ei, ea, amat, uvec, vvec, amax, denom,
                                                    hl, h_out, E);
        h_in = h_out;
    }
    // h_in == d_out after layer 2; reshape [B, N/B, HN] is a no-op on flat memory.
}